// S4_6425271075222
// MI455X (gfx1250) — compile-verified
//
#include <hip/hip_runtime.h>

#define BATCH   32
#define SEQL    336
#define HDIM    256
#define NSEQ    862
#define DOUTN   336
#define NLAYERS 4
#define NMODES  32
#define NT      54   // ceil(NSEQ/16)

typedef float v2f __attribute__((ext_vector_type(2)));
typedef float v8f __attribute__((ext_vector_type(8)));

__device__ __forceinline__ v8f wmma4(v2f a, v2f b, v8f c) {
  // V_WMMA_F32_16X16X4_F32 : D = A(16x4) * B(4x16) + C (all f32)
  return __builtin_amdgcn_wmma_f32_16x16x4_f32(
      false, a, false, b, (short)0, c, false, false);
}

__device__ __forceinline__ float gelu_exact(float v) {
  return 0.5f * v * (1.0f + erff(v * 0.70710678118654752f));
}

__device__ __forceinline__ int imin(int a, int b) { return a < b ? a : b; }
__device__ __forceinline__ int imax(int a, int b) { return a > b ? a : b; }

// ---------------------------------------------------------------------------
// S4D discretization: Cp = (C_re + i C_im) * (exp(dt*A)-1)/A ; dA = dt*A
// P layout: [Cpr | Cpi | dAr | dAi], each H*NMODES
// ---------------------------------------------------------------------------
__global__ __launch_bounds__(256) void s4_prep(
    const float* __restrict__ log_dt, const float* __restrict__ C_re,
    const float* __restrict__ C_im, const float* __restrict__ log_A_real,
    const float* __restrict__ A_imag, float* __restrict__ P) {
  int idx = blockIdx.x * 256 + threadIdx.x;
  if (idx >= HDIM * NMODES) return;
  int h = idx / NMODES;
  float dt  = expf(log_dt[h]);
  float Ar  = -expf(log_A_real[idx]);
  float Ai  = A_imag[idx];
  float dAr = Ar * dt, dAi = Ai * dt;
  float ex  = expf(dAr);
  float s, c;
  sincosf(dAi, &s, &c);
  float er = ex * c - 1.0f, ei = ex * s;
  float den = Ar * Ar + Ai * Ai;
  float fr = (er * Ar + ei * Ai) / den;
  float fi = (ei * Ar - er * Ai) / den;
  float cr = C_re[idx], ci = C_im[idx];
  P[idx]                     = cr * fr - ci * fi;
  P[HDIM * NMODES + idx]     = cr * fi + ci * fr;
  P[2 * HDIM * NMODES + idx] = dAr;
  P[3 * HDIM * NMODES + idx] = dAi;
}

// ---------------------------------------------------------------------------
// Vandermonde kernel: K[h,l] = 2 * Re sum_n Cp[h,n] * exp(dA[h,n] * l)
// ---------------------------------------------------------------------------
__global__ __launch_bounds__(256) void s4_vander(const float* __restrict__ P,
                                                 float* __restrict__ Kk) {
  __shared__ float sCpr[NMODES], sCpi[NMODES], sAr[NMODES], sAi[NMODES];
  int h = blockIdx.x;
  int t = threadIdx.x;
  if (t < NMODES) {
    sCpr[t] = P[h * NMODES + t];
    sCpi[t] = P[HDIM * NMODES + h * NMODES + t];
    sAr[t]  = P[2 * HDIM * NMODES + h * NMODES + t];
    sAi[t]  = P[3 * HDIM * NMODES + h * NMODES + t];
  }
  __syncthreads();
  for (int l = t; l < NSEQ; l += 256) {
    float lf = (float)l;
    float acc = 0.0f;
#pragma unroll
    for (int n = 0; n < NMODES; ++n) {
      float er = expf(sAr[n] * lf);
      float s, c;
      sincosf(sAi[n] * lf, &s, &c);
      acc += sCpr[n] * (er * c) - sCpi[n] * (er * s);
    }
    Kk[(size_t)h * NSEQ + l] = 2.0f * acc;
  }
}

// ---------------------------------------------------------------------------
// Causal convolution y = K * u as triangular Toeplitz GEMM per channel h,
// fused + D[h]*u and exact GELU.  One wave per (l-tile, h); M = 32 batch rows
// (two 16-row WMMA tiles sharing the B tile), K-dim = shift m.
// Main loop (m < l0) is provably in-range -> unconditional loads.
// Diagonal tail (4 iters) uses branchless clamp + 0/1 mask.
// ---------------------------------------------------------------------------
__global__ __launch_bounds__(32) void s4_conv(
    const float* __restrict__ x, const float* __restrict__ Kk,
    const float* __restrict__ Dv, float* __restrict__ y) {
  int lt = blockIdx.x, h = blockIdx.y;
  int lane = threadIdx.x;
  int row = lane & 15, half = lane >> 4;
  int l0 = lt * 16;
  int pos  = l0 + row;
  int posc = imin(pos, NSEQ - 1);  // clamped: dead columns read valid junk
  const float* Krow = Kk + (size_t)h * NSEQ;
  const float* xr0 = x + ((size_t)row * HDIM + h) * NSEQ;         // b = row
  const float* xr1 = x + ((size_t)(row + 16) * HDIM + h) * NSEQ;  // b = row+16
  v8f acc0 = {0.f, 0.f, 0.f, 0.f, 0.f, 0.f, 0.f, 0.f};
  v8f acc1 = acc0;
  // ---- main body: m in [0, l0), all indices in range ----
  for (int m = 0; m < l0; m += 4) {
    int k0 = m + 2 * half;
    v2f a0 = *(const v2f*)(xr0 + k0);
    v2f a1 = *(const v2f*)(xr1 + k0);
    int ki = posc - k0;  // >= 1 here
    v2f bt;
    bt.x = Krow[ki];
    bt.y = Krow[ki - 1];
    acc0 = wmma4(a0, bt, acc0);
    acc1 = wmma4(a1, bt, acc1);
  }
  // ---- diagonal tail: m in [l0, l0+16), branchless masking ----
#pragma unroll
  for (int mt = 0; mt < 4; ++mt) {
    int m = l0 + mt * 4;
    int k0 = m + 2 * half;
    int kc = imin(k0, NSEQ - 2);  // keep v2f load in range (last tile only)
    v2f av0 = *(const v2f*)(xr0 + kc);
    v2f av1 = *(const v2f*)(xr1 + kc);
    float mk0 = (k0 < NSEQ) ? 1.f : 0.f;
    float mk1 = (k0 + 1 < NSEQ) ? 1.f : 0.f;
    v2f a0, a1, bt;
    a0.x = av0.x * mk0; a0.y = av0.y * mk1;
    a1.x = av1.x * mk0; a1.y = av1.y * mk1;
    int ki = posc - k0;
    float bx = Krow[imax(ki, 0)];
    float by = Krow[imax(ki - 1, 0)];
    bt.x = (ki >= 0) ? bx : 0.f;
    bt.y = (ki >= 1) ? by : 0.f;
    acc0 = wmma4(a0, bt, acc0);
    acc1 = wmma4(a1, bt, acc1);
  }
  float Dh = Dv[h];
  if (pos < NSEQ) {  // C/D: n = lane&15, m = vgpr + 8*(lane>>4)
#pragma unroll
    for (int r = 0; r < 8; ++r) {
      int mrow = r + 8 * half;
      size_t i0 = ((size_t)mrow * HDIM + h) * NSEQ + pos;
      y[i0] = gelu_exact(acc0[r] + Dh * x[i0]);
      size_t i1 = ((size_t)(mrow + 16) * HDIM + h) * NSEQ + pos;
      y[i1] = gelu_exact(acc1[r] + Dh * x[i1]);
    }
  }
}

// ---------------------------------------------------------------------------
// 1x1 conv (2H x H) computing both GLU halves (shared B tile), fused
// a*sigmoid(b) + residual.  Unconditional loads via clamped column index.
// ---------------------------------------------------------------------------
__global__ __launch_bounds__(32) void s4_mix_glu(
    const float* __restrict__ y, const float* __restrict__ w,
    const float* __restrict__ bvec, const float* __restrict__ xres,
    float* __restrict__ z) {
  int nt = blockIdx.x, ot = blockIdx.y, b = blockIdx.z;
  int lane = threadIdx.x;
  int row = lane & 15, half = lane >> 4;
  int n0 = nt * 16, o0 = ot * 16;
  int pos  = n0 + row;
  int posc = imin(pos, NSEQ - 1);
  const float* wa = w + (size_t)(o0 + row) * HDIM;        // GLU "a" rows
  const float* wb = w + (size_t)(o0 + 256 + row) * HDIM;  // GLU "b" rows
  const float* yb = y + (size_t)b * HDIM * NSEQ + posc;
  v8f accA = {0.f, 0.f, 0.f, 0.f, 0.f, 0.f, 0.f, 0.f};
  v8f accB = accA;
#pragma unroll 4
  for (int k = 0; k < HDIM; k += 4) {
    int k0 = k + 2 * half;
    v2f aa = *(const v2f*)(wa + k0);
    v2f ab = *(const v2f*)(wb + k0);
    v2f bt;
    bt.x = yb[(size_t)k0 * NSEQ];
    bt.y = yb[(size_t)(k0 + 1) * NSEQ];
    accA = wmma4(aa, bt, accA);
    accB = wmma4(ab, bt, accB);
  }
  if (pos < NSEQ) {
#pragma unroll
    for (int r = 0; r < 8; ++r) {
      int o = o0 + r + 8 * half;
      float av = accA[r] + bvec[o];
      float bv = accB[r] + bvec[256 + o];
      float glu = av * (1.0f / (1.0f + expf(-bv)));
      size_t idx = ((size_t)b * HDIM + o) * NSEQ + pos;
      z[idx] = glu + xres[idx];
    }
  }
}

// ---------------------------------------------------------------------------
// In-place LayerNorm over H for each (b, n)
// ---------------------------------------------------------------------------
__global__ __launch_bounds__(256) void ln_kernel(float* __restrict__ z,
                                                 const float* __restrict__ g,
                                                 const float* __restrict__ be) {
  __shared__ float s1[256], s2[256];
  int bn = blockIdx.x;
  int b = bn / NSEQ, n = bn % NSEQ;
  int h = threadIdx.x;
  size_t idx = ((size_t)b * HDIM + h) * NSEQ + n;
  float v = z[idx];
  s1[h] = v;
  s2[h] = v * v;
  __syncthreads();
  for (int s = 128; s > 0; s >>= 1) {
    if (h < s) {
      s1[h] += s1[h + s];
      s2[h] += s2[h + s];
    }
    __syncthreads();
  }
  float mu = s1[0] * (1.0f / HDIM);
  float var = s2[0] * (1.0f / HDIM) - mu * mu;
  z[idx] = (v - mu) * rsqrtf(var + 1e-5f) * g[h] + be[h];
}

// ---------------------------------------------------------------------------
// Encoder: x[b,h,n] = sum_l enc_w[h,l] * hist[b,l,n] + enc_b[h]
// Two 16-row h-tiles per wave (M=32) sharing the B tile.
// ---------------------------------------------------------------------------
__global__ __launch_bounds__(32) void enc_gemm(
    const float* __restrict__ hist, const float* __restrict__ w,
    const float* __restrict__ bias, float* __restrict__ x) {
  int nt = blockIdx.x, ht = blockIdx.y, b = blockIdx.z;
  int lane = threadIdx.x;
  int row = lane & 15, half = lane >> 4;
  int n0 = nt * 16, h0 = ht * 32;
  int pos  = n0 + row;
  int posc = imin(pos, NSEQ - 1);
  const float* wr0 = w + (size_t)(h0 + row) * SEQL;
  const float* wr1 = w + (size_t)(h0 + 16 + row) * SEQL;
  const float* hb = hist + (size_t)b * SEQL * NSEQ + posc;
  v8f acc0 = {0.f, 0.f, 0.f, 0.f, 0.f, 0.f, 0.f, 0.f};
  v8f acc1 = acc0;
#pragma unroll 4
  for (int k = 0; k < SEQL; k += 4) {
    int k0 = k + 2 * half;
    v2f a0 = *(const v2f*)(wr0 + k0);
    v2f a1 = *(const v2f*)(wr1 + k0);
    v2f bt;
    bt.x = hb[(size_t)k0 * NSEQ];
    bt.y = hb[(size_t)(k0 + 1) * NSEQ];
    acc0 = wmma4(a0, bt, acc0);
    acc1 = wmma4(a1, bt, acc1);
  }
  if (pos < NSEQ) {
#pragma unroll
    for (int r = 0; r < 8; ++r) {
      int hh = h0 + r + 8 * half;
      x[((size_t)b * HDIM + hh) * NSEQ + pos] = acc0[r] + bias[hh];
      x[((size_t)b * HDIM + hh + 16) * NSEQ + pos] = acc1[r] + bias[hh + 16];
    }
  }
}

// ---------------------------------------------------------------------------
// Decoder: out[b,o,n] = sum_h dec_w[o,h] * x[b,h,n] + dec_b[o]
// Two 16-col n-tiles per wave (N=32) sharing the A tile.
// ---------------------------------------------------------------------------
__global__ __launch_bounds__(32) void dec_gemm(
    const float* __restrict__ x, const float* __restrict__ w,
    const float* __restrict__ bias, float* __restrict__ out) {
  int nt = blockIdx.x, ot = blockIdx.y, b = blockIdx.z;
  int lane = threadIdx.x;
  int row = lane & 15, half = lane >> 4;
  int n0 = nt * 32, o0 = ot * 16;
  int pos0 = n0 + row, pos1 = n0 + 16 + row;
  int pc0 = imin(pos0, NSEQ - 1), pc1 = imin(pos1, NSEQ - 1);
  const float* wr = w + (size_t)(o0 + row) * HDIM;
  const float* xb0 = x + (size_t)b * HDIM * NSEQ + pc0;
  const float* xb1 = x + (size_t)b * HDIM * NSEQ + pc1;
  v8f acc0 = {0.f, 0.f, 0.f, 0.f, 0.f, 0.f, 0.f, 0.f};
  v8f acc1 = acc0;
#pragma unroll 4
  for (int k = 0; k < HDIM; k += 4) {
    int k0 = k + 2 * half;
    v2f aa = *(const v2f*)(wr + k0);
    v2f b0, b1;
    b0.x = xb0[(size_t)k0 * NSEQ];
    b0.y = xb0[(size_t)(k0 + 1) * NSEQ];
    b1.x = xb1[(size_t)k0 * NSEQ];
    b1.y = xb1[(size_t)(k0 + 1) * NSEQ];
    acc0 = wmma4(aa, b0, acc0);
    acc1 = wmma4(aa, b1, acc1);
  }
#pragma unroll
  for (int r = 0; r < 8; ++r) {
    int o = o0 + r + 8 * half;
    float bo = bias[o];
    if (pos0 < NSEQ)
      out[((size_t)b * DOUTN + o) * NSEQ + pos0] = acc0[r] + bo;
    if (pos1 < NSEQ)
      out[((size_t)b * DOUTN + o) * NSEQ + pos1] = acc1[r] + bo;
  }
}

// ---------------------------------------------------------------------------
extern "C" void kernel_launch(void* const* d_in, const int* in_sizes, int n_in,
                              void* d_out, int out_size, void* d_ws,
                              size_t ws_size, hipStream_t stream) {
  const float* hist       = (const float*)d_in[0];
  const float* enc_w      = (const float*)d_in[5];
  const float* enc_b      = (const float*)d_in[6];
  const float* dec_w      = (const float*)d_in[7];
  const float* dec_b      = (const float*)d_in[8];
  const float* log_dt     = (const float*)d_in[9];
  const float* C_re       = (const float*)d_in[10];
  const float* C_im       = (const float*)d_in[11];
  const float* log_A_real = (const float*)d_in[12];
  const float* A_imag     = (const float*)d_in[13];
  const float* Dv         = (const float*)d_in[14];
  const float* out_w      = (const float*)d_in[15];
  const float* out_b      = (const float*)d_in[16];
  const float* ln_g       = (const float*)d_in[17];
  const float* ln_b       = (const float*)d_in[18];
  float* out = (float*)d_out;

  const size_t BHN = (size_t)BATCH * HDIM * NSEQ;
  float* bufA = (float*)d_ws;       // x ping
  float* bufB = bufA + BHN;         // x pong
  float* bufY = bufB + BHN;         // gelu(conv) output
  float* Kk   = bufY + BHN;         // H * NSEQ conv kernel
  float* P    = Kk + (size_t)HDIM * NSEQ;  // 4 * H * NMODES coefficients

  enc_gemm<<<dim3(NT, HDIM / 32, BATCH), 32, 0, stream>>>(hist, enc_w, enc_b,
                                                          bufA);

  for (int i = 0; i < NLAYERS; ++i) {
    float* cur = (i & 1) ? bufB : bufA;
    float* nxt = (i & 1) ? bufA : bufB;
    s4_prep<<<(HDIM * NMODES + 255) / 256, 256, 0, stream>>>(
        log_dt + i * HDIM, C_re + i * HDIM * NMODES, C_im + i * HDIM * NMODES,
        log_A_real + i * HDIM * NMODES, A_imag + i * HDIM * NMODES, P);
    s4_vander<<<HDIM, 256, 0, stream>>>(P, Kk);
    s4_conv<<<dim3(NT, HDIM), 32, 0, stream>>>(cur, Kk, Dv + i * HDIM, bufY);
    s4_mix_glu<<<dim3(NT, HDIM / 16, BATCH), 32, 0, stream>>>(
        bufY, out_w + (size_t)i * 2 * HDIM * HDIM, out_b + i * 2 * HDIM, cur,
        nxt);
    ln_kernel<<<BATCH * NSEQ, 256, 0, stream>>>(nxt, ln_g + i * HDIM,
                                                ln_b + i * HDIM);
  }

  // after 4 layers output is in bufA (0:A->B, 1:B->A, 2:A->B, 3:B->A)
  dec_gemm<<<dim3(NT / 2, DOUTN / 16, BATCH), 32, 0, stream>>>(bufA, dec_w,
                                                               dec_b, out);
}